// GATAspects_15307263443308
// MI455X (gfx1250) — compile-verified
//
#include <hip/hip_runtime.h>
#include <math.h>

// Problem constants from the reference: N nodes, D neighbors, F features.
#define NN 10000
#define DD 16
#define FF 256

typedef __attribute__((ext_vector_type(2))) float v2f;
typedef __attribute__((ext_vector_type(4))) float v4f;
typedef __attribute__((ext_vector_type(8))) float v8f;

// ---------------------------------------------------------------------------
// K1: fold all the small linear maps into score vectors.
//   v  = Wa^T a_src            (2F)
//   u1 = Wp^T v[:F], u2 = Wp^T v[F:], ut = Wp^T a_tgt   (F each)
//   c  = ba . a_src
// ---------------------------------------------------------------------------
__global__ __launch_bounds__(256) void k_precompute(
    const float* __restrict__ Wp, const float* __restrict__ Wa,
    const float* __restrict__ ba, const float* __restrict__ a_src,
    const float* __restrict__ a_tgt, float* __restrict__ ws_u) {
  __shared__ float sv[2 * FF];
  __shared__ float sa[FF];
  const int t = threadIdx.x;
  sa[t] = a_src[t];
  __syncthreads();

  float v0 = 0.f, v1 = 0.f;
  for (int f = 0; f < FF; ++f) {
    const float as = sa[f];
    v0 += Wa[f * (2 * FF) + t] * as;
    v1 += Wa[f * (2 * FF) + t + FF] * as;
  }
  sv[t] = v0;
  sv[t + FF] = v1;
  __syncthreads();

  float u1 = 0.f, u2 = 0.f, ut = 0.f;
  for (int g = 0; g < FF; ++g) {
    const float w = Wp[g * FF + t];  // Wp[g, t] : coalesced across threads
    u1 += w * sv[g];
    u2 += w * sv[FF + g];
    ut += w * a_tgt[g];
  }
  ws_u[t] = u1;
  ws_u[FF + t] = u2;
  ws_u[2 * FF + t] = ut;
  if (t == 0) {
    float c = 0.f;
    for (int f = 0; f < FF; ++f) c += ba[f] * sa[f];
    ws_u[3 * FF] = c;
  }
}

// ---------------------------------------------------------------------------
// K2: one workgroup (8 wave32s) per node. Bandwidth-critical kernel:
//   - float4 (b128) non-temporal streaming of neighbors/aspects (touched once;
//     keep the 328 MB stream out of L2 so Wp/agg stay resident)
//   - neighbors staged in LDS (16 KB of the 320 KB WGP budget) for reuse
//   - wave32 shuffle reductions for the D=16 source scores + target score
//   - raw-exp softmax over D (matches reference: no max subtraction)
//   - attention-weighted aggregate of raw neighbor features -> ws agg[n, :]
// ---------------------------------------------------------------------------
__global__ __launch_bounds__(256) void k_node_scores_agg(
    const float* __restrict__ nodes, const float* __restrict__ neigh,
    const float* __restrict__ asp, const float* __restrict__ ws_u,
    float* __restrict__ agg) {
  __shared__ __align__(16) float sn[DD * FF];  // 16 KB neighbor tile
  __shared__ float s_sc[DD];
  __shared__ float s_attn[DD];

  const int n = blockIdx.x;
  const int t = threadIdx.x;
  const int wave = t >> 5;
  const int lane = t & 31;

  const v4f* __restrict__ u14 = (const v4f*)ws_u;
  const v4f* __restrict__ u24 = (const v4f*)(ws_u + FF);
  const v4f* __restrict__ ut4 = (const v4f*)(ws_u + 2 * FF);
  const float cadd = ws_u[3 * FF];

  const v4f* __restrict__ nb4 = (const v4f*)(neigh + (size_t)n * (DD * FF));
  const v4f* __restrict__ ab4 = (const v4f*)(asp + (size_t)n * (DD * FF));
  v4f* __restrict__ sn4 = (v4f*)sn;

  // Each wave owns d = wave and d = wave + 8. Row = 64 float4s, 2 per lane.
  #pragma unroll
  for (int dd = 0; dd < 2; ++dd) {
    const int d = wave + dd * 8;
    float ps = 0.f;
    #pragma unroll
    for (int j = 0; j < 2; ++j) {
      const int idx = lane + j * 32;  // float4 index within the row
      const v4f nv = __builtin_nontemporal_load(&nb4[d * 64 + idx]);
      const v4f av = __builtin_nontemporal_load(&ab4[d * 64 + idx]);
      const v4f w1 = u14[idx];
      const v4f w2 = u24[idx];
      sn4[d * 64 + idx] = nv;  // stage for aggregation
      ps += nv.x * w1.x + nv.y * w1.y + nv.z * w1.z + nv.w * w1.w;
      ps += av.x * w2.x + av.y * w2.y + av.z * w2.z + av.w * w2.w;
    }
    #pragma unroll
    for (int m = 16; m >= 1; m >>= 1) ps += __shfl_xor(ps, m, 32);
    if (lane == 0) s_sc[d] = ps + cadd;
  }

  // Wave 0 additionally computes the target score (broadcast to all lanes).
  float st = 0.f;
  if (wave == 0) {
    const v4f* __restrict__ nd4 = (const v4f*)(nodes + (size_t)n * FF);
    #pragma unroll
    for (int j = 0; j < 2; ++j) {
      const int idx = lane + j * 32;
      const v4f xv = __builtin_nontemporal_load(&nd4[idx]);
      const v4f wt = ut4[idx];
      st += xv.x * wt.x + xv.y * wt.y + xv.z * wt.z + xv.w * wt.w;
    }
    #pragma unroll
    for (int m = 16; m >= 1; m >>= 1) st += __shfl_xor(st, m, 32);
  }
  __syncthreads();

  // Wave 0: leaky-relu, raw exp, normalize over D=16 (lanes 16..31 mirror 0..15).
  if (wave == 0) {
    const int d = lane & 15;
    float s = s_sc[d] + st;
    s = (s > 0.f) ? s : 0.2f * s;
    const float e = __expf(s);
    float sum = e;
    #pragma unroll
    for (int m = 8; m >= 1; m >>= 1) sum += __shfl_xor(sum, m, 32);
    if (lane < 16) s_attn[d] = e / (sum + 1e-16f);
  }
  __syncthreads();

  // Aggregate raw neighbor features (Wp projection deferred to the WMMA GEMM).
  float a = 0.f;
  #pragma unroll
  for (int d = 0; d < DD; ++d) a += s_attn[d] * sn[d * FF + t];
  agg[(size_t)n * FF + t] = a;
}

// ---------------------------------------------------------------------------
// K3: out = ELU(agg @ Wp^T + bias) via V_WMMA_F32_16X16X4_F32.
//   One block per M-tile (625 blocks); wave w owns N-tiles {2w, 2w+1} and
//   shares one A-fragment across both -> 2 WMMAs per A load, agg re-read 8x
//   (L2-resident) instead of 16x. K-loop: 64 steps of K=4.
//   B[k, col] = Wp[col, k], so B-frag loads are rows of Wp (L2-resident).
// ---------------------------------------------------------------------------
__global__ __launch_bounds__(256) void k_wmma_gemm_elu(
    const float* __restrict__ agg, const float* __restrict__ Wp,
    const float* __restrict__ bias, float* __restrict__ out) {
  const int t = threadIdx.x;
  const int lane = t & 31;
  const int wave = t >> 5;
  const int mtile = blockIdx.x;      // 0..624
  const int ntile0 = wave * 2;       // 0,2,..,14
  const int ntile1 = ntile0 + 1;

  const int r = lane & 15;           // A: row M, B: col N (per ISA layout)
  const int kb = (lane >> 4) << 1;   // K sub-pair: lanes 0-15 -> {k,k+1}, 16-31 -> {k+2,k+3}

  const float* __restrict__ Ap = agg + (size_t)(mtile * 16 + r) * FF + kb;
  const float* __restrict__ Bp0 = Wp + (size_t)(ntile0 * 16 + r) * FF + kb;
  const float* __restrict__ Bp1 = Bp0 + 16 * FF;

  v8f c0 = {};
  v8f c1 = {};
  #pragma unroll 4
  for (int k = 0; k < FF; k += 4) {
    const v2f a = *(const v2f*)(Ap + k);
    const v2f b0 = *(const v2f*)(Bp0 + k);
    const v2f b1 = *(const v2f*)(Bp1 + k);
    c0 = __builtin_amdgcn_wmma_f32_16x16x4_f32(false, a, false, b0, (short)0,
                                               c0, false, false);
    c1 = __builtin_amdgcn_wmma_f32_16x16x4_f32(false, a, false, b1, (short)0,
                                               c1, false, false);
  }

  // C/D layout: VGPR i -> M = i + 8*(lane>=16), N = lane & 15.
  const int mb = mtile * 16 + ((lane >> 4) << 3);
  const int nc0 = ntile0 * 16 + r;
  const int nc1 = ntile1 * 16 + r;
  const float bn0 = bias[nc0];
  const float bn1 = bias[nc1];
  #pragma unroll
  for (int i = 0; i < 8; ++i) {
    float v0 = c0[i] + bn0;
    float v1 = c1[i] + bn1;
    v0 = (v0 > 0.f) ? v0 : (__expf(v0) - 1.f);  // ELU, alpha = 1
    v1 = (v1 > 0.f) ? v1 : (__expf(v1) - 1.f);
    __builtin_nontemporal_store(v0, &out[(size_t)(mb + i) * FF + nc0]);
    __builtin_nontemporal_store(v1, &out[(size_t)(mb + i) * FF + nc1]);
  }
}

// ---------------------------------------------------------------------------
// Workspace layout (floats):
//   [0, 256)      u1
//   [256, 512)    u2
//   [512, 768)    ut
//   [768]         c = ba . a_src
//   [1024, ...)   agg[N, F]  (10.24 MB)
// ---------------------------------------------------------------------------
extern "C" void kernel_launch(void* const* d_in, const int* in_sizes, int n_in,
                              void* d_out, int out_size, void* d_ws,
                              size_t ws_size, hipStream_t stream) {
  const float* nodes = (const float*)d_in[0];
  const float* neigh = (const float*)d_in[1];
  const float* asp   = (const float*)d_in[2];
  const float* Wp    = (const float*)d_in[3];
  const float* Wa    = (const float*)d_in[4];
  const float* ba    = (const float*)d_in[5];
  const float* a_src = (const float*)d_in[6];
  const float* a_tgt = (const float*)d_in[7];
  const float* bias  = (const float*)d_in[8];
  float* out = (float*)d_out;
  float* wsf = (float*)d_ws;
  float* agg = wsf + 1024;

  k_precompute<<<1, 256, 0, stream>>>(Wp, Wa, ba, a_src, a_tgt, wsf);
  k_node_scores_agg<<<NN, 256, 0, stream>>>(nodes, neigh, asp, wsf, agg);
  k_wmma_gemm_elu<<<NN / 16, 256, 0, stream>>>(agg, Wp, bias, out);
}